// CharEmbeddingV03x01_4063039062450
// MI455X (gfx1250) — compile-verified
//
#include <hip/hip_runtime.h>
#include <math.h>

// CDNA5 / gfx1250. wave32. One fused kernel:
//  Phase 0: stage Wout into zero-padded LDS tile [40][16] (K-pad, N-pad).
//  Phase 1: each of 256 threads evaluates the full tiny-MLP pipeline for one
//           class id (the whole network is a function of id only) and writes
//           its 38 "cat" features (padded to 40) to LDS.
//  Phase 2: 8 waves run the 256x40 @ 40x7 final dense with
//           V_WMMA_F32_16X16X4_F32 (f32, full precision, unconditional LDS
//           fragment loads -> clean back-to-back WMMA chain), add bias, tanh,
//           store the 256x7 table transposed into LDS.
//  Phase 3: grid-stride gather: out[t*7+j] = tbl[j][ids[t]]  (HBM roofline:
//           ~17 MB total traffic => ~0.7 us at 23.3 TB/s).
//
// WMMA f32 16x16x4 fragment layout (ISA 7.12.2):
//   A (16x4): VGPR0 = K0 (lanes0-15) / K2 (lanes16-31); VGPR1 = K1 / K3.
//   B (4x16): same K pattern, N = lane%16.
//   D (16x16, 8 VGPRs): VGPR v -> M = v (lanes0-15) / v+8 (lanes16-31), N = lane%16.

typedef __attribute__((ext_vector_type(2))) float v2f;
typedef __attribute__((ext_vector_type(8))) float v8f;

#define CSTR 41   // cat_lds row stride (odd -> conflict-free A-fragment reads)

__global__ __launch_bounds__(256)
void char_emb_lut_kernel(const int* __restrict__ ids,
                         const float* __restrict__ emb,
                         const float* __restrict__ W72, const float* __restrict__ b72,
                         const float* __restrict__ W73, const float* __restrict__ b73,
                         const float* __restrict__ W75, const float* __restrict__ b75,
                         const float* __restrict__ W21, const float* __restrict__ b21,
                         const float* __restrict__ W31, const float* __restrict__ b31,
                         const float* __restrict__ W51, const float* __restrict__ b51,
                         const float* __restrict__ W71, const float* __restrict__ b71,
                         const float* __restrict__ Wout, const float* __restrict__ bout,
                         float* __restrict__ out, int ntok) {
    __shared__ float cat_lds[256 * CSTR];   // 256 classes x 40 (padded K) feats
    __shared__ float wout_lds[40 * 16];     // [k][n], zero-padded B operand
    __shared__ float tbl[7 * 256];          // transposed table: tbl[j*256 + id]

    const int tid = threadIdx.x;            // 0..255 == class id in phase 1

    // ---------------- Phase 0: stage zero-padded Wout ------------------------
    for (int i = tid; i < 40 * 16; i += 256) {
        const int k = i >> 4;
        const int n = i & 15;
        wout_lds[i] = (n < 7 && k < 38) ? Wout[k * 7 + n] : 0.0f;
    }

    // ---------------- Phase 1: per-class feature pipeline (scalar) ----------
    {
        const int c = tid;
        float e[7];
#pragma unroll
        for (int j = 0; j < 7; ++j) e[j] = emb[c * 7 + j];

        float d72[2], d73[3], d75[5];
#pragma unroll
        for (int j = 0; j < 2; ++j) {
            float a = b72[j];
#pragma unroll
            for (int i = 0; i < 7; ++i) a += e[i] * W72[i * 2 + j];
            d72[j] = tanhf(a);
        }
#pragma unroll
        for (int j = 0; j < 3; ++j) {
            float a = b73[j];
#pragma unroll
            for (int i = 0; i < 7; ++i) a += e[i] * W73[i * 3 + j];
            d73[j] = tanhf(a);
        }
#pragma unroll
        for (int j = 0; j < 5; ++j) {
            float a = b75[j];
#pragma unroll
            for (int i = 0; i < 7; ++i) a += e[i] * W75[i * 5 + j];
            d75[j] = tanhf(a);
        }
        float a21 = b21[0];
#pragma unroll
        for (int i = 0; i < 2; ++i) a21 += d72[i] * W21[i];
        float d21 = tanhf(a21);
        float a31 = b31[0];
#pragma unroll
        for (int i = 0; i < 3; ++i) a31 += d73[i] * W31[i];
        float d31 = tanhf(a31);
        float a51 = b51[0];
#pragma unroll
        for (int i = 0; i < 5; ++i) a51 += d75[i] * W51[i];
        float d51 = tanhf(a51);
        float a71 = b71[0];
#pragma unroll
        for (int i = 0; i < 7; ++i) a71 += e[i] * W71[i];
        float d71 = tanhf(a71);

        float* row = &cat_lds[c * CSTR];
#pragma unroll
        for (int j = 0; j < 7; ++j) row[j] = e[j];
#pragma unroll
        for (int j = 0; j < 3; ++j) row[7 + j] = d73[j];
        row[10] = d71; row[11] = d51; row[12] = d31; row[13] = d21;
        // pairs in reference order: (21,31)(21,51)(21,71)(31,51)(31,71)(51,71)
        const float pa[6] = {d21, d21, d21, d31, d31, d51};
        const float pb[6] = {d31, d51, d71, d51, d71, d71};
#pragma unroll
        for (int p = 0; p < 6; ++p) {
            row[14 + p] = pa[p] * pb[p];
            row[20 + p] = pa[p] + pb[p];
            row[26 + p] = fmaxf(pa[p], pb[p]);
            row[32 + p] = fminf(pa[p], pb[p]);
        }
        row[38] = 0.0f;   // K padding 38..39
        row[39] = 0.0f;
    }
    __syncthreads();

    // ---------------- Phase 2: 256x40 @ 40x7 via V_WMMA_F32_16X16X4_F32 -----
    {
        const int wave = tid >> 5;
        const int lane = tid & 31;
        const int half = lane >> 4;       // 0: lanes 0-15, 1: lanes 16-31
        const int lrow = lane & 15;       // M row / N col within tile

        const float bo = (lrow < 7) ? bout[lrow] : 0.0f;

        // Preload the 10 B fragments (tile-invariant) from padded LDS: no
        // divergence, no exec masking -> clean WMMA chain.
        v2f Bf[10];
#pragma unroll
        for (int kk = 0; kk < 10; ++kk) {
            const int k0 = kk * 4 + half * 2;
            Bf[kk].x = wout_lds[(k0    ) * 16 + lrow];
            Bf[kk].y = wout_lds[(k0 + 1) * 16 + lrow];
        }

#pragma unroll
        for (int tile = 0; tile < 2; ++tile) {
            const int rowbase = wave * 32 + tile * 16;
            const float* arow = &cat_lds[(rowbase + lrow) * CSTR];
            v8f acc = {};
#pragma unroll
            for (int kk = 0; kk < 10; ++kk) {
                const int k0 = kk * 4 + half * 2;       // this lane's K pair
                v2f A; A.x = arow[k0]; A.y = arow[k0 + 1];
                acc = __builtin_amdgcn_wmma_f32_16x16x4_f32(
                    false, A, false, Bf[kk], (short)0, acc, false, false);
            }
            if (lrow < 7) {
#pragma unroll
                for (int v = 0; v < 8; ++v) {
                    const int r = rowbase + v + half * 8;   // class id 0..255
                    tbl[lrow * 256 + r] = tanhf(acc[v] + bo);
                }
            }
        }
    }
    __syncthreads();

    // ---------------- Phase 3: roofline gather -------------------------------
    const int stride = gridDim.x * 256;
    for (int t = blockIdx.x * 256 + tid; t < ntok; t += stride) {
        const int id = ids[t];
        float* o = out + (size_t)t * 7;
#pragma unroll
        for (int j = 0; j < 7; ++j) o[j] = tbl[j * 256 + id];
    }
}

extern "C" void kernel_launch(void* const* d_in, const int* in_sizes, int n_in,
                              void* d_out, int out_size, void* d_ws, size_t ws_size,
                              hipStream_t stream) {
    const int*   ids  = (const int*)  d_in[0];
    const float* emb  = (const float*)d_in[1];
    const float* W72  = (const float*)d_in[2];  const float* b72 = (const float*)d_in[3];
    const float* W73  = (const float*)d_in[4];  const float* b73 = (const float*)d_in[5];
    const float* W75  = (const float*)d_in[6];  const float* b75 = (const float*)d_in[7];
    const float* W21  = (const float*)d_in[8];  const float* b21 = (const float*)d_in[9];
    const float* W31  = (const float*)d_in[10]; const float* b31 = (const float*)d_in[11];
    const float* W51  = (const float*)d_in[12]; const float* b51 = (const float*)d_in[13];
    const float* W71  = (const float*)d_in[14]; const float* b71 = (const float*)d_in[15];
    const float* Wout = (const float*)d_in[16]; const float* bout= (const float*)d_in[17];
    float* out = (float*)d_out;

    const int ntok = in_sizes[0];                 // B*L = 524288
    int blocks = (ntok + 255) / 256;
    if (blocks > 2048) blocks = 2048;
    if (blocks < 1) blocks = 1;

    char_emb_lut_kernel<<<blocks, 256, 0, stream>>>(
        ids, emb, W72, b72, W73, b73, W75, b75,
        W21, b21, W31, b31, W51, b51, W71, b71,
        Wout, bout, out, ntok);
}